// Attention_77446850281941
// MI455X (gfx1250) — compile-verified
//
#include <hip/hip_runtime.h>
#include <hip/hip_bf16.h>
#include <cmath>

// ---------------------------------------------------------------------------
// Attention scores + softmax for MI455X (gfx1250).
//
// Roofline: enc_outputs is 256 MB (f32) and is touched exactly once ->
// ~11 us floor at 23.3 TB/s HBM. Everything else (dec dot, softmax passes)
// is KB-to-sub-MB. So the design is a streaming GEMV over a 65536x1024
// row-major matrix, using V_WMMA_F32_16X16X4_F32 so the dot products ride
// the matrix pipe in full f32 precision (compute is free under the memory
// roof). B (= w_enc) is broadcast across all 16 WMMA columns so every
// column of D carries the same 16 dot products.
// ---------------------------------------------------------------------------

typedef __attribute__((ext_vector_type(2))) float v2f;
typedef __attribute__((ext_vector_type(4))) float v4f;
typedef __attribute__((ext_vector_type(8))) float v8f;

#define BATCH   32
#define SRC_LEN 2048
#define K_DIM   1024                 // 2*enc_hid_dim == dec_hid_dim == 1024
#define ROWS    (SRC_LEN * BATCH)    // 65536 rows, row r = s*32 + b

// ---------------------------------------------------------------------------
// Kernel 1: e_dec[b] = dec_hidden[b,:] . W[0,:1024]      (32 dots of len 1024)
// One wave per batch element; wave32 shuffle reduction.
// ---------------------------------------------------------------------------
__global__ __launch_bounds__(1024) void edec_kernel(const float* __restrict__ dh,
                                                    const float* __restrict__ W,
                                                    float* __restrict__ edec) {
    const int wave = threadIdx.x >> 5;   // 32 waves == 32 batch rows
    const int lane = threadIdx.x & 31;
    const float* row = dh + (size_t)wave * K_DIM;
    float acc = 0.0f;
    for (int i = lane; i < K_DIM; i += 32)
        acc = fmaf(row[i], W[i], acc);
    for (int off = 16; off > 0; off >>= 1)
        acc += __shfl_down(acc, off, 32);
    if (lane == 0) edec[wave] = acc;
}

// ---------------------------------------------------------------------------
// Kernel 2: the 256 MB stream. One wave owns 16 consecutive rows (a 16x1024
// tile). Per 8-element k-block: one global b128 per lane (lanes 0-15 take
// k..k+3 of their row, lanes 16-31 take k+4..k+7 -> 32B contiguous per row
// per instruction, every byte of each row consumed), one LDS b128 broadcast
// of w_enc, and two v_wmma_f32_16x16x4_f32. The k->WMMA-slot mapping is a
// permutation applied identically to A and B, so the contraction is exact.
// Two accumulators break the serial WMMA RAW chain.
// Stores exp(tanh(score) - 1): shift-invariant softmax numerator (tanh <= 1).
// ---------------------------------------------------------------------------
__global__ __launch_bounds__(256) void score_kernel(const float* __restrict__ enc,
                                                    const float* __restrict__ W,
                                                    const float* __restrict__ bias,
                                                    const float* __restrict__ edec,
                                                    float* __restrict__ out) {
    __shared__ __align__(16) float lds_w[K_DIM];  // w_enc staged once per WG (4 KB)
    __shared__ float dots[8][16];                 // per-wave dot extraction bounce

    // cooperative coalesced stage of w_enc = W[0, 1024:2048]
    const float* w_enc = W + K_DIM;
    for (int i = threadIdx.x; i < K_DIM; i += 256)
        lds_w[i] = w_enc[i];
    __syncthreads();

    const int wave = threadIdx.x >> 5;
    const int lane = threadIdx.x & 31;
    const int tile = blockIdx.x * 8 + wave;   // 512 blocks * 8 waves = 4096 tiles
    const int r0   = tile * 16;
    const int hi   = lane >> 4;               // 0: lanes 0-15, 1: lanes 16-31
    const int m    = lane & 15;               // matrix row within tile

    const float* rowp = enc + (size_t)(r0 + m) * K_DIM + hi * 4;
    const float* wp   = lds_w + hi * 4;

    v8f c0 = {};
    v8f c1 = {};

    #pragma unroll 4
    for (int k0 = 0; k0 < K_DIM; k0 += 16) {
        v4f a0 = *(const v4f*)(rowp + k0);        // global b128, row-major stream
        v4f a1 = *(const v4f*)(rowp + k0 + 8);
        v4f w0 = *(const v4f*)(wp + k0);          // LDS b128, same-address broadcast
        v4f w1 = *(const v4f*)(wp + k0 + 8);
        // (neg_a, A, neg_b, B, c_mod, C, reuse_a, reuse_b)
        c0 = __builtin_amdgcn_wmma_f32_16x16x4_f32(false, a0.xy, false, w0.xy,
                                                   (short)0, c0, false, false);
        c0 = __builtin_amdgcn_wmma_f32_16x16x4_f32(false, a0.zw, false, w0.zw,
                                                   (short)0, c0, false, false);
        c1 = __builtin_amdgcn_wmma_f32_16x16x4_f32(false, a1.xy, false, w1.xy,
                                                   (short)0, c1, false, false);
        c1 = __builtin_amdgcn_wmma_f32_16x16x4_f32(false, a1.zw, false, w1.zw,
                                                   (short)0, c1, false, false);
    }
    c0 = c0 + c1;

    // C/D layout: VGPR j, lanes 0-15 -> D[j, lane]; lanes 16-31 -> D[j+8, lane-16].
    // All columns equal (B broadcast), so lane 0 holds rows 0..7, lane 16 rows 8..15.
    if (lane == 0) {
        #pragma unroll
        for (int j = 0; j < 8; ++j) dots[wave][j] = c0[j];
    } else if (lane == 16) {
        #pragma unroll
        for (int j = 0; j < 8; ++j) dots[wave][8 + j] = c0[j];
    }
    __syncthreads();

    if (lane < 16) {
        const int row = r0 + lane;       // r = s*32 + b
        const int b   = row & 31;
        const int s   = row >> 5;
        const float e = dots[wave][lane] + edec[b] + bias[0];
        const float t = tanhf(e);
        out[(size_t)b * SRC_LEN + s] = __expf(t - 1.0f);   // softmax numerator
    }
}

// ---------------------------------------------------------------------------
// Kernel 3: per-batch normalization. out[b, :] /= sum(out[b, :]).
// 32 blocks x 256 threads over 2048 elements each; trivial vs the 256 MB pass.
// ---------------------------------------------------------------------------
__global__ __launch_bounds__(256) void softmax_norm_kernel(float* __restrict__ out) {
    __shared__ float red[256];
    float* row = out + (size_t)blockIdx.x * SRC_LEN;
    float s = 0.0f;
    for (int i = threadIdx.x; i < SRC_LEN; i += 256)
        s += row[i];
    red[threadIdx.x] = s;
    __syncthreads();
    for (int off = 128; off > 0; off >>= 1) {
        if (threadIdx.x < off) red[threadIdx.x] += red[threadIdx.x + off];
        __syncthreads();
    }
    const float inv = 1.0f / red[0];
    for (int i = threadIdx.x; i < SRC_LEN; i += 256)
        row[i] *= inv;
}

// ---------------------------------------------------------------------------
extern "C" void kernel_launch(void* const* d_in, const int* in_sizes, int n_in,
                              void* d_out, int out_size, void* d_ws, size_t ws_size,
                              hipStream_t stream) {
    const float* dec_hidden = (const float*)d_in[0];  // (32, 1024) f32
    const float* enc_out    = (const float*)d_in[1];  // (2048, 32, 1024) f32
    const float* W          = (const float*)d_in[2];  // (1, 2048) f32
    const float* bias       = (const float*)d_in[3];  // (1,) f32
    float* out  = (float*)d_out;                      // (32, 2048) f32
    float* edec = (float*)d_ws;                       // 32 floats scratch

    edec_kernel<<<1, 1024, 0, stream>>>(dec_hidden, W, edec);
    score_kernel<<<ROWS / 16 / 8, 256, 0, stream>>>(enc_out, W, bias, edec, out);
    softmax_norm_kernel<<<BATCH, 256, 0, stream>>>(out);
}